// HistogramLoss_15040975470954
// MI455X (gfx1250) — compile-verified
//
#include <hip/hip_runtime.h>

#define NUM_BINS 256
#define WAVES_PER_BLOCK 8
#define HIST_SLOTS (2 * NUM_BINS)   // input bins [0..255], target bins [256..511]

typedef __attribute__((ext_vector_type(4))) float f4;
typedef __attribute__((ext_vector_type(2))) float v2f;
typedef __attribute__((ext_vector_type(8))) float v8f;

// ---------------------------------------------------------------------------
// Kernel 0: zero the 512-entry global histogram in workspace (ws is poisoned
// with 0xAA by the harness and never re-poisoned between replays).
// ---------------------------------------------------------------------------
__global__ void hist_zero_kernel(unsigned int* __restrict__ gh) {
    int i = blockIdx.x * blockDim.x + threadIdx.x;
    if (i < HIST_SLOTS) gh[i] = 0u;
}

// ---------------------------------------------------------------------------
// Kernel 1: streaming histogram. Per-wave private LDS histograms (wave32:
// 8 waves/block, zero cross-wave LDS contention), float4 NT loads.
// torch.histc semantics: values outside [0,255] ignored; x==255 -> last bin.
// ---------------------------------------------------------------------------
__device__ __forceinline__ void bin_one(float x, unsigned int* __restrict__ h, int base) {
    if (x >= 0.0f && x <= 255.0f) {
        // x >= 0 so int-cast truncation == floor
        int b = (int)(x * (256.0f / 255.0f));
        b = b > (NUM_BINS - 1) ? (NUM_BINS - 1) : b;
        atomicAdd(&h[base + b], 1u);
    }
}

__global__ void __launch_bounds__(256)
hist_kernel(const float* __restrict__ inp, const float* __restrict__ tgt,
            unsigned int* __restrict__ gh, long long n) {
    __shared__ unsigned int lh[WAVES_PER_BLOCK * HIST_SLOTS];  // 16 KB

    // zero LDS histograms
    for (int j = threadIdx.x; j < WAVES_PER_BLOCK * HIST_SLOTS; j += blockDim.x)
        lh[j] = 0u;
    __syncthreads();

    unsigned int* h = &lh[(threadIdx.x >> 5) * HIST_SLOTS];  // per-wave private

    const f4* __restrict__ in4 = (const f4*)inp;
    const f4* __restrict__ tg4 = (const f4*)tgt;
    long long n4     = n >> 2;
    long long stride = (long long)gridDim.x * blockDim.x;

    for (long long i = (long long)blockIdx.x * blockDim.x + threadIdx.x; i < n4; i += stride) {
        f4 a = __builtin_nontemporal_load(&in4[i]);   // global_load_b128, TH=NT
        f4 b = __builtin_nontemporal_load(&tg4[i]);
#pragma unroll
        for (int k = 0; k < 4; ++k) {
            bin_one(a[k], h, 0);
            bin_one(b[k], h, NUM_BINS);
        }
    }

    // scalar tail (n is a multiple of 4 for this shape, but stay correct)
    if (blockIdx.x == 0 && threadIdx.x == 0) {
        for (long long i = (n & ~3LL); i < n; ++i) {
            bin_one(inp[i], h, 0);
            bin_one(tgt[i], h, NUM_BINS);
        }
    }

    __syncthreads();
    // merge the 8 wave-private histograms, flush once per block to global
    for (int j = threadIdx.x; j < HIST_SLOTS; j += blockDim.x) {
        unsigned int s = 0;
#pragma unroll
        for (int w = 0; w < WAVES_PER_BLOCK; ++w) s += lh[w * HIST_SLOTS + j];
        if (s) atomicAdd(&gh[j], s);
    }
}

// ---------------------------------------------------------------------------
// Layout-independent 256-value sum via V_WMMA_F32_16X16X4_F32.
// A = all-ones(16x4), B = 64 values (arbitrary slot mapping), C-chained over
// 4 groups. Every row of D is the column-sum vector, so
//   sum(all 256 D entries) = 16 * sum(values)
// regardless of the exact A/B/D VGPR layouts. All 32 lanes of every wave
// execute (EXEC all-ones, as WMMA requires); result broadcast via shuffles.
// ---------------------------------------------------------------------------
__device__ __forceinline__ float sum256_wmma(float v, float* __restrict__ buf, int t) {
    __syncthreads();           // buf may be in use from a previous call
    buf[t] = v;
    __syncthreads();

    int lane = t & 31;
    v2f A; A[0] = 1.0f; A[1] = 1.0f;
    v8f D = {};
#pragma unroll
    for (int g = 0; g < 4; ++g) {
        v2f B;
        B[0] = buf[g * 64 + 2 * lane + 0];
        B[1] = buf[g * 64 + 2 * lane + 1];
        // (neg_a, A, neg_b, B, c_mod, C, reuse_a, reuse_b)
        D = __builtin_amdgcn_wmma_f32_16x16x4_f32(false, A, false, B,
                                                  (short)0, D, false, false);
    }
    float s = 0.0f;
#pragma unroll
    for (int k = 0; k < 8; ++k) s += D[k];
    for (int off = 16; off > 0; off >>= 1) s += __shfl_down(s, off, 32);
    s = __shfl(s, 0, 32);      // broadcast within wave (every wave computed it)
    return s * (1.0f / 16.0f);
}

// ---------------------------------------------------------------------------
// Kernel 2: normalize histograms, intersection = sum(min), out = 1 - inter.
// One block, 256 threads (one per bin).
// ---------------------------------------------------------------------------
__global__ void __launch_bounds__(256)
finalize_kernel(const unsigned int* __restrict__ gh, float* __restrict__ out) {
    __shared__ float buf[NUM_BINS];
    int t = threadIdx.x;

    float ci = (float)gh[t];             // per-bin counts are ~196K << 2^24: exact in f32
    float ti = (float)gh[NUM_BINS + t];

    float Si = sum256_wmma(ci, buf, t);
    float St = sum256_wmma(ti, buf, t);

    float m = fminf(ci / Si, ti / St);
    float inter = sum256_wmma(m, buf, t);

    if (t == 0) out[0] = 1.0f - inter;
}

// ---------------------------------------------------------------------------
extern "C" void kernel_launch(void* const* d_in, const int* in_sizes, int n_in,
                              void* d_out, int out_size, void* d_ws, size_t ws_size,
                              hipStream_t stream) {
    const float* inp = (const float*)d_in[0];
    const float* tgt = (const float*)d_in[1];
    float* out       = (float*)d_out;
    unsigned int* gh = (unsigned int*)d_ws;  // 512 uints

    long long n = (long long)in_sizes[0];

    hist_zero_kernel<<<2, 256, 0, stream>>>(gh);
    // 1536 blocks x 256 threads: 393,216 threads -> exactly 32 float4
    // iterations per thread per array for n = 50,331,648.
    hist_kernel<<<1536, 256, 0, stream>>>(inp, tgt, gh, n);
    finalize_kernel<<<1, 256, 0, stream>>>(gh, out);
}